// LocalInfoNCELoss_30906584662479
// MI455X (gfx1250) — compile-verified
//
#include <hip/hip_runtime.h>
#include <hip/hip_bf16.h>

typedef float v2f __attribute__((ext_vector_type(2)));
typedef float v8f __attribute__((ext_vector_type(8)));

#define BS 16
#define Hh 192
#define Ww 192
#define Cc 64
#define KK 9          // 3x3 cells per region
#define R_ 13
#define TWO_R 26
#define D_ 576        // KK*Cc
#define ROWS_PAD 32
#define TAU_INV 2.0f  // 1/0.5
#define EPSV 1e-8f

#define PN_ELEMS (BS * ROWS_PAD * D_)   // 294912 floats
// ws layout: [pn: PN_ELEMS floats][sim: BS*ROWS_PAD*ROWS_PAD floats]

// --- Kernel 1: gather + row L2-normalize into padded pn (BS,32,576) ---------
__global__ __launch_bounds__(192)
void k_gather_norm(const float* __restrict__ f1, const float* __restrict__ f2,
                   const int* __restrict__ b_idx, const int* __restrict__ h_idx,
                   const int* __restrict__ w_idx, float* __restrict__ pn)
{
    int blk = blockIdx.x;          // 0 .. BS*ROWS_PAD-1
    int b = blk >> 5;
    int j = blk & 31;
    int t = threadIdx.x;           // 0..191, 3 elems each (3*192 = 576)
    float* out = pn + (size_t)blk * D_;

    if (j >= TWO_R) {              // zero-pad rows 26..31 for clean WMMA tiles
        out[t] = 0.f; out[t + 192] = 0.f; out[t + 384] = 0.f;
        return;
    }
    const float* f = (j < R_) ? f1 : f2;
    int r = (j < R_) ? j : (j - R_);

    float v[3];
#pragma unroll
    for (int e = 0; e < 3; ++e) {
        int d = t + e * 192;
        int k = d >> 6;            // cell within region
        int c = d & 63;            // channel
        int i = (r * BS + b) * KK + k;   // flat gather index (region-major)
        int bi = b_idx[i], hi = h_idx[i], wi = w_idx[i];
        v[e] = f[(((size_t)bi * Hh + hi) * Ww + wi) * Cc + c];
    }
    float ss = v[0] * v[0] + v[1] * v[1] + v[2] * v[2];

    __shared__ float red[192];
    red[t] = ss;
    __syncthreads();
    if (t < 64) red[t] += red[t + 64] + red[t + 128];   // 192 -> 64
    __syncthreads();
    for (int s = 32; s > 0; s >>= 1) {
        if (t < s) red[t] += red[t + s];
        __syncthreads();
    }
    float inv = 1.0f / fmaxf(sqrtf(red[0]), EPSV);
    out[t]       = v[0] * inv;
    out[t + 192] = v[1] * inv;
    out[t + 384] = v[2] * inv;
}

// --- Kernel 2: per-batch Gram matrix sim = (pn pn^T)/tau via f32 WMMA -------
// One block per batch, 4 waves = 4 16x16 tiles of the padded 32x32 Gram.
__global__ __launch_bounds__(128)
void k_gram_wmma(const float* __restrict__ pn, float* __restrict__ sim)
{
    int b    = blockIdx.x;             // 0..15
    int wave = threadIdx.x >> 5;       // 0..3
    int lane = threadIdx.x & 31;
    int ti = wave >> 1, tj = wave & 1;
    int m16 = lane & 15;
    int kh  = (lane >> 4) << 1;        // K sub-offset: 0 (lanes 0-15) / 2 (16-31)

    const float* base = pn + (size_t)b * ROWS_PAD * D_;
    const float* arow = base + (size_t)(ti * 16 + m16) * D_ + kh;  // A: 16x4 tile rows
    const float* brow = base + (size_t)(tj * 16 + m16) * D_ + kh;  // B: same layout (Gram)

    v8f acc = {0.f, 0.f, 0.f, 0.f, 0.f, 0.f, 0.f, 0.f};
    for (int kk = 0; kk < D_; kk += 4) {
        v2f a  = *(const v2f*)(arow + kk);
        v2f bb = *(const v2f*)(brow + kk);
        // D = A(16x4) * B(4x16) + C, f32 end to end (matches reference numerics)
        acc = __builtin_amdgcn_wmma_f32_16x16x4_f32(false, a, false, bb,
                                                    (short)0, acc, false, false);
    }

    // C/D layout: VGPR r, lanes 0-15 -> M=r, lanes 16-31 -> M=r+8; N = lane%16
    float* s = sim + (size_t)b * ROWS_PAD * ROWS_PAD;
    int n  = tj * 16 + m16;
    int mb = ti * 16 + ((lane >> 4) << 3);
#pragma unroll
    for (int r2 = 0; r2 < 8; ++r2)
        s[(mb + r2) * ROWS_PAD + n] = acc[r2] * TAU_INV;
}

// --- Kernel 3: InfoNCE loss. Row loss = LSE_{j!=i}(sim[i,j]) - sim[i,(i+R)%2R]
__global__ __launch_bounds__(512)
void k_loss(const float* __restrict__ sim, float* __restrict__ out)
{
    int t = threadIdx.x;
    float loss = 0.f;
    if (t < BS * TWO_R) {
        int b = t / TWO_R;
        int i = t % TWO_R;
        const float* row = sim + ((size_t)b * ROWS_PAD + i) * ROWS_PAD;
        float m = -1e30f;
        for (int j = 0; j < TWO_R; ++j)
            if (j != i) m = fmaxf(m, row[j]);
        float se = 0.f;
        for (int j = 0; j < TWO_R; ++j)
            if (j != i) se += expf(row[j] - m);
        float pos = row[(i + R_) % TWO_R];
        loss = logf(se) + m - pos;
    }
    __shared__ float red[512];
    red[t] = loss;
    __syncthreads();
    for (int s = 256; s > 0; s >>= 1) {
        if (t < s) red[t] += red[t + s];
        __syncthreads();
    }
    if (t == 0) out[0] = red[0] / (float)(BS * TWO_R);
}

extern "C" void kernel_launch(void* const* d_in, const int* in_sizes, int n_in,
                              void* d_out, int out_size, void* d_ws, size_t ws_size,
                              hipStream_t stream)
{
    const float* f1   = (const float*)d_in[0];
    const float* f2   = (const float*)d_in[1];
    const int* b_idx  = (const int*)d_in[2];
    const int* h_idx  = (const int*)d_in[3];
    const int* w_idx  = (const int*)d_in[4];

    float* pn  = (float*)d_ws;
    float* sim = pn + PN_ELEMS;

    k_gather_norm<<<BS * ROWS_PAD, 192, 0, stream>>>(f1, f2, b_idx, h_idx, w_idx, pn);
    k_gram_wmma<<<BS, 128, 0, stream>>>(pn, sim);
    k_loss<<<1, 512, 0, stream>>>(sim, (float*)d_out);
}